// TCNResBlock_32719060861384
// MI455X (gfx1250) — compile-verified
//
#include <hip/hip_runtime.h>
#include <math.h>

// Reference shapes (fixed by setup_inputs): B=32, C=384, T=8192, K=3, dilation=1.
// Memory-bound (AI ~1.9 F/B): strategy = 1 workgroup per (b,c) row, row DMA'd
// into LDS via gfx1250 async global->LDS b128, two-pass norm + 3-tap stencil
// entirely from LDS, streaming non-temporal b128 stores.

typedef float v4f __attribute__((ext_vector_type(4)));

#define AS3 __attribute__((address_space(3)))

constexpr int B_ = 32;
constexpr int C_ = 384;
constexpr int T_ = 8192;
constexpr int THREADS = 256;                 // 8 wave32s
constexpr int PER_THREAD = T_ / THREADS;     // 32 elements / thread
constexpr int V4_PER_THREAD = PER_THREAD / 4; // 8 float4s / thread
constexpr float EPS_ = 1e-5f;

// Workgroup-relative LDS byte address of a generic pointer into __shared__.
__device__ __forceinline__ unsigned lds_byte_addr(const void* p) {
  return (unsigned)(unsigned long long)(AS3 const void*)p;
}

// gfx1250 async DMA: 16 bytes per lane, global -> LDS, tracked by ASYNCcnt.
__device__ __forceinline__ void async_load_b128(unsigned lds_off, const void* gaddr) {
  asm volatile("global_load_async_to_lds_b128 %0, %1, off"
               :
               : "v"(lds_off), "v"(gaddr)
               : "memory");
}

__device__ __forceinline__ void wait_asynccnt0() {
#if defined(__has_builtin)
#if __has_builtin(__builtin_amdgcn_s_wait_asynccnt)
  __builtin_amdgcn_s_wait_asynccnt(0);
  return;
#endif
#endif
  asm volatile("s_wait_asynccnt 0" ::: "memory");
}

__global__ __launch_bounds__(THREADS)
void tcn_resblock_kernel(const float* __restrict__ inp,
                         const float* __restrict__ weight,  // [C,1,3]
                         const float* __restrict__ bias,    // [C]
                         float* __restrict__ out) {
  __shared__ float s_in[T_];          // raw input row (also residual source)
  __shared__ float s_elu[T_ + 2];     // ELU(xhat), shifted by 1; halos = 0
  __shared__ float s_psum[THREADS / 32];
  __shared__ float s_psq[THREADS / 32];

  const int row = blockIdx.x;         // row = b*C + c
  const int c   = row % C_;
  const int tid = threadIdx.x;

  const float* row_in = inp + (size_t)row * T_;

  // ---- Pass 0: async-DMA the whole 32KB row into LDS ------------------
  const unsigned lds_in_base = lds_byte_addr(&s_in[0]);
#pragma unroll
  for (int k = 0; k < T_ / (4 * THREADS); ++k) {     // 8 b128 issues / thread
    const int j = k * THREADS + tid;                 // float4 index in row
    async_load_b128(lds_in_base + (unsigned)j * 16u,
                    (const void*)(row_in + (size_t)j * 4));
  }
  wait_asynccnt0();
  __syncthreads();

  // ---- Pass 1: mean / variance over T (wave32 butterfly + LDS partials)
  float sum = 0.f, ssq = 0.f;
  const int t0 = tid * PER_THREAD;
#pragma unroll
  for (int k = 0; k < V4_PER_THREAD; ++k) {
    v4f v = *(const v4f*)&s_in[t0 + k * 4];          // 16B-aligned ds_load_b128
    sum += (v.x + v.y) + (v.z + v.w);
    ssq += (v.x * v.x + v.y * v.y) + (v.z * v.z + v.w * v.w);
  }
#pragma unroll
  for (int off = 16; off > 0; off >>= 1) {
    sum += __shfl_xor(sum, off, 32);
    ssq += __shfl_xor(ssq, off, 32);
  }
  if ((tid & 31) == 0) {
    s_psum[tid >> 5] = sum;
    s_psq[tid >> 5]  = ssq;
  }
  __syncthreads();
  float tot = 0.f, totsq = 0.f;
#pragma unroll
  for (int w = 0; w < THREADS / 32; ++w) {           // redundant per-thread sum
    tot   += s_psum[w];
    totsq += s_psq[w];
  }
  const float mean = tot * (1.f / (float)T_);
  const float var  = totsq * (1.f / (float)T_) - mean * mean;  // biased, as jnp.var
  const float inv  = rsqrtf(var + EPS_);

  // ---- Pass 2: normalize + ELU into shifted buffer (zero-padded halos)
  if (tid == 0) { s_elu[0] = 0.f; s_elu[T_ + 1] = 0.f; }
#pragma unroll
  for (int k = 0; k < V4_PER_THREAD; ++k) {
    v4f v = *(const v4f*)&s_in[t0 + k * 4];
#pragma unroll
    for (int i = 0; i < 4; ++i) {
      const float x = (v[i] - mean) * inv;
      s_elu[1 + t0 + k * 4 + i] = (x > 0.f) ? x : expm1f(x);  // ELU, alpha=1
    }
  }
  __syncthreads();

  // ---- Pass 3: depthwise conv (cross-correlation) + bias + residual ----
  const float w0 = weight[c * 3 + 0];
  const float w1 = weight[c * 3 + 1];
  const float w2 = weight[c * 3 + 2];
  const float bs = bias[c];

  float* row_out = out + (size_t)row * T_;
  float em1 = s_elu[t0];          // e[t0-1] (or 0 at left edge)
  float e0  = s_elu[t0 + 1];      // e[t0]
#pragma unroll
  for (int k = 0; k < V4_PER_THREAD; ++k) {
    v4f r = *(const v4f*)&s_in[t0 + k * 4];          // residual
    v4f y;
#pragma unroll
    for (int i = 0; i < 4; ++i) {
      const float ep1 = s_elu[t0 + k * 4 + i + 2];   // e[t+1] (or 0 at right edge)
      y[i] = fmaf(w0, em1, fmaf(w1, e0, fmaf(w2, ep1, bs))) + r[i];
      em1 = e0;
      e0  = ep1;
    }
    // Output is write-once streaming: non-temporal b128 store.
    __builtin_nontemporal_store(y, (v4f*)(row_out + t0 + k * 4));
  }
}

extern "C" void kernel_launch(void* const* d_in, const int* in_sizes, int n_in,
                              void* d_out, int out_size, void* d_ws, size_t ws_size,
                              hipStream_t stream) {
  (void)in_sizes; (void)n_in; (void)out_size; (void)d_ws; (void)ws_size;
  const float* inp    = (const float*)d_in[0];
  const float* weight = (const float*)d_in[1];
  const float* bias   = (const float*)d_in[2];
  float* out          = (float*)d_out;

  dim3 grid(B_ * C_);      // 12288 rows, one workgroup each
  dim3 block(THREADS);
  hipLaunchKernelGGL(tcn_resblock_kernel, grid, block, 0, stream,
                     inp, weight, bias, out);
}